// LocalContrastLayer_56805237457216
// MI455X (gfx1250) — compile-verified
//
#include <hip/hip_runtime.h>

// ---------------------------------------------------------------------------
// LocalContrastLayer: 6x6 box mean / mean-of-squares, normalize, sigmoid(2z)
//   out = 0.5*(tanh((x-mu)/sigma)+1) == 1/(1+exp(-2*(x-mu)/sigma))
// 128 x 512 x 512 x 1 fp32. Pure HBM-bandwidth-bound (~268 MB min traffic).
// Tile 32x128 outputs -> 37x136 input halo tile (read amplification 1.23x).
// Input staged via GLOBAL_LOAD_ASYNC_TO_LDS_B128 (ASYNCcnt-tracked DMA path).
// ---------------------------------------------------------------------------

namespace {
constexpr int IMG_H = 512;
constexpr int IMG_W = 512;
constexpr int TH    = 32;            // output rows per block
constexpr int TW    = 128;           // output cols per block
constexpr int IN_H  = TH + 5;        // 37 input rows (pad lo=2, hi=3)
constexpr int SW    = TW + 8;        // 136 floats/LDS row; halo [-4,+4) keeps float4 global alignment
constexpr int HS_W  = 132;           // 132 % 64 == 4 -> conflict-free hs writes
constexpr int BLOCK = 256;
constexpr float EPS = 1e-5f;
}

// ---- async global->LDS path (signature verified by compiler) --------------
#if defined(__has_builtin)
#if __has_builtin(__builtin_amdgcn_global_load_async_to_lds_b128)
#define LCN_HAVE_ASYNC 1
#endif
#if __has_builtin(__builtin_amdgcn_s_wait_asynccnt)
#define LCN_HAVE_WAIT_ASYNC 1
#endif
#endif

#ifdef LCN_HAVE_ASYNC
typedef int lcn_v4i __attribute__((vector_size(16)));
typedef __attribute__((address_space(1))) lcn_v4i* lcn_gptr;   // global (non-const per builtin proto)
typedef __attribute__((address_space(3))) lcn_v4i* lcn_lptr;   // LDS

__device__ __forceinline__ void lcn_async_load16(const float* g, float* l) {
  __builtin_amdgcn_global_load_async_to_lds_b128(
      (lcn_gptr)g, (lcn_lptr)l, /*offset=*/0, /*cpol=*/0);
}
__device__ __forceinline__ void lcn_wait_async0() {
#ifdef LCN_HAVE_WAIT_ASYNC
  __builtin_amdgcn_s_wait_asynccnt(0);
#else
  asm volatile("s_wait_asynccnt 0" ::: "memory");
#endif
}
#endif

__global__ __launch_bounds__(BLOCK)
void lcn_kernel(const float* __restrict__ in, float* __restrict__ out) {
  __shared__ __align__(16) float sx[IN_H * SW];     // raw input tile (+halo)   ~20.1 KB
  __shared__ float hs [IN_H * HS_W];                // horizontal 6-sum of x    ~19.5 KB
  __shared__ float hs2[IN_H * HS_W];                // horizontal 6-sum of x^2  ~19.5 KB

  const int t  = threadIdx.x;
  const int c0 = blockIdx.x * TW;                   // output col origin
  const int r0 = blockIdx.y * TH;                   // output row origin
  const size_t img_off = (size_t)blockIdx.z * (size_t)(IMG_H * IMG_W);
  const float* __restrict__ img  = in  + img_off;
  float* __restrict__       oimg = out + img_off;

  // ---- Stage 0: zero-fill LDS tile only when the halo leaves the image ----
  const bool border = (r0 == 0) | (r0 + TH == IMG_H) | (c0 == 0) | (c0 + TW == IMG_W);
  if (border) {
    const float4 z = make_float4(0.f, 0.f, 0.f, 0.f);
    for (int e = t; e < IN_H * (SW / 4); e += BLOCK)
      ((float4*)sx)[e] = z;
    __syncthreads();
  }

  // ---- Stage 1: stage 37x136 input tile into LDS (float4 granules) -------
  // LDS col j <-> global col (c0 - 4 + j); LDS row i <-> global row (r0 - 2 + i).
  // Every float4 granule is fully in-image or fully out (W % 4 == 0, origin % 4 == 0).
  for (int e = t; e < IN_H * (SW / 4); e += BLOCK) {
    const int r  = e / (SW / 4);
    const int q  = e - r * (SW / 4);
    const int gr = r0 - 2 + r;
    const int gc = c0 - 4 + q * 4;
    if ((unsigned)gr < (unsigned)IMG_H && (unsigned)gc < (unsigned)IMG_W) {
      const float* gp = img + (size_t)gr * IMG_W + gc;
      float*       lp = sx + r * SW + q * 4;
#ifdef LCN_HAVE_ASYNC
      lcn_async_load16(gp, lp);
#else
      *(float4*)lp = *(const float4*)gp;
#endif
    }
  }
#ifdef LCN_HAVE_ASYNC
  lcn_wait_async0();          // per-wave drain of ASYNCcnt before the barrier
#endif
  __syncthreads();

  // ---- Stage 2: horizontal running 6-sums (x and x^2) ---------------------
  // 37 rows x 6 threads; each thread owns a 22-col segment (last gets 18).
  // Output col c uses input cols c-2..c+3  ->  LDS cols c+2..c+7.
  if (t < IN_H * 6) {
    const int i    = t / 6;
    const int k    = t - i * 6;
    const int cbeg = k * 22;
    const int cend = (cbeg + 22 < TW) ? (cbeg + 22) : TW;
    const float* row = sx + i * SW;
    float s1 = 0.f, s2 = 0.f;
#pragma unroll
    for (int j = 0; j < 5; ++j) {
      const float v = row[cbeg + 2 + j];
      s1 += v; s2 += v * v;
    }
    for (int c = cbeg; c < cend; ++c) {
      const float v = row[c + 7];
      s1 += v; s2 += v * v;
      hs [i * HS_W + c] = s1;
      hs2[i * HS_W + c] = s2;
      const float u = row[c + 2];
      s1 -= u; s2 -= u * u;
    }
  }
  __syncthreads();

  // ---- Stage 3: vertical running 6-sums + epilogue + store ----------------
  // 128 cols x 2 row-halves; thread owns 16 output rows of one column.
  // Output row r uses hs rows r..r+5 (max index 36, in bounds).
  {
    const int c  = t & (TW - 1);
    const int rb = (t >> 7) * (TH / 2);
    float s1 = 0.f, s2 = 0.f;
#pragma unroll
    for (int i = 0; i < 6; ++i) {
      s1 += hs [(rb + i) * HS_W + c];
      s2 += hs2[(rb + i) * HS_W + c];
    }
#pragma unroll
    for (int k = 0; k < TH / 2; ++k) {
      const int r = rb + k;
      const float m   = s1 * (1.0f / 36.0f);
      const float mq  = s2 * (1.0f / 36.0f);
      const float var = fmaxf(mq - m * m, 0.0f) + EPS;
      const float x   = sx[(r + 2) * SW + (c + 4)];
      const float z   = (x - m) * rsqrtf(var);
      // 0.5*(tanh(z)+1) == 1/(1+exp(-2z))
      const float o   = 1.0f / (1.0f + __expf(-2.0f * z));
      oimg[(size_t)(r0 + r) * IMG_W + (c0 + c)] = o;
      if (k < TH / 2 - 1) {
        s1 += hs [(r + 6) * HS_W + c] - hs [r * HS_W + c];
        s2 += hs2[(r + 6) * HS_W + c] - hs2[r * HS_W + c];
      }
    }
  }
}

extern "C" void kernel_launch(void* const* d_in, const int* in_sizes, int n_in,
                              void* d_out, int out_size, void* d_ws, size_t ws_size,
                              hipStream_t stream) {
  (void)n_in; (void)out_size; (void)d_ws; (void)ws_size;
  const float* in = (const float*)d_in[0];
  float* out = (float*)d_out;
  const int n_img = in_sizes[0] / (IMG_H * IMG_W);   // 128
  dim3 grid(IMG_W / TW, IMG_H / TH, n_img);          // 4 x 16 x 128
  lcn_kernel<<<grid, dim3(BLOCK), 0, stream>>>(in, out);
}